// SSDLoss_54760833024677
// MI455X (gfx1250) — compile-verified
//
#include <hip/hip_runtime.h>
#include <hip/hip_bf16.h>

#define NUM_CLASSES 21
#define NUM_PRIORS  8732
#define BATCH       128
#define NUM_GT      20
#define NTHREADS    512

typedef __attribute__((ext_vector_type(2))) float v2f;
typedef __attribute__((ext_vector_type(8))) float v8f;

__device__ __forceinline__ float sl1(float x) {
    float ax = fabsf(x);
    return ax < 1.0f ? 0.5f * x * x : ax - 0.5f;
}

__global__ __launch_bounds__(NTHREADS)
void ssd_match_kernel(const float* __restrict__ loc,
                      const float* __restrict__ conf,
                      const float* __restrict__ defbox,
                      const float* __restrict__ gt_boxes,
                      const int*   __restrict__ gt_labels,
                      float* __restrict__ wsLoss,
                      int*   __restrict__ wsNpos)
{
    __shared__ float              sIou[NUM_PRIORS];
    __shared__ unsigned char      sIdx[NUM_PRIORS];
    __shared__ float              sNeg[NUM_PRIORS];   // loss_c (0 for positives)
    __shared__ float              sGt[NUM_GT * 4];
    __shared__ int                sGl[NUM_GT];
    __shared__ unsigned long long sBestP[NUM_GT];
    __shared__ float              sRedF[NTHREADS];
    __shared__ int                sRedI[NTHREADS];
    __shared__ int                sHist[256];
    __shared__ unsigned           sPrefix;
    __shared__ int                sRemain;

    const int b   = blockIdx.x;
    const int tid = threadIdx.x;

    // ---- Stage GT boxes into LDS with CDNA5 async-to-LDS copy ----
    if (tid < NUM_GT) {
        unsigned lds_off = (unsigned)(unsigned long long)(&sGt[tid * 4]);
        unsigned long long ga =
            (unsigned long long)(gt_boxes + ((size_t)b * NUM_GT + tid) * 4);
        asm volatile("global_load_async_to_lds_b128 %0, %1, off"
                     :: "v"(lds_off), "v"(ga) : "memory");
        sGl[tid]    = gt_labels[b * NUM_GT + tid];
        sBestP[tid] = 0ull;
        asm volatile("s_wait_asynccnt 0" ::: "memory");
    }
    __syncthreads();

    // ---- Pass 1: per-prior best IoU / argmax over GTs (first-occurrence ties) ----
    for (int p = tid; p < NUM_PRIORS; p += NTHREADS) {
        float dcx = defbox[p * 4 + 0], dcy = defbox[p * 4 + 1];
        float dw  = defbox[p * 4 + 2], dh  = defbox[p * 4 + 3];
        float ax0 = dcx - 0.5f * dw, ay0 = dcy - 0.5f * dh;
        float ax1 = dcx + 0.5f * dw, ay1 = dcy + 0.5f * dh;
        float areaA = dw * dh;
        float best = -1.0f; int bidx = 0;
        for (int g = 0; g < NUM_GT; ++g) {
            float bx0 = sGt[g * 4 + 0], by0 = sGt[g * 4 + 1];
            float bx1 = sGt[g * 4 + 2], by1 = sGt[g * 4 + 3];
            float iw = fmaxf(fminf(ax1, bx1) - fmaxf(ax0, bx0), 0.0f);
            float ih = fmaxf(fminf(ay1, by1) - fmaxf(ay0, by0), 0.0f);
            float inter = iw * ih;
            float areaB = (bx1 - bx0) * (by1 - by0);
            float iou = inter / (areaA + areaB - inter);
            if (iou > best) { best = iou; bidx = g; }
        }
        sIou[p] = best;
        sIdx[p] = (unsigned char)bidx;
    }

    // ---- Pass 2: per-GT argmax over priors via packed u64 LDS max (deterministic) ----
    for (int g = 0; g < NUM_GT; ++g) {
        float bx0 = sGt[g * 4 + 0], by0 = sGt[g * 4 + 1];
        float bx1 = sGt[g * 4 + 2], by1 = sGt[g * 4 + 3];
        float areaB = (bx1 - bx0) * (by1 - by0);
        float best = -1.0f; int bp = 0;
        for (int p = tid; p < NUM_PRIORS; p += NTHREADS) {
            float dcx = defbox[p * 4 + 0], dcy = defbox[p * 4 + 1];
            float dw  = defbox[p * 4 + 2], dh  = defbox[p * 4 + 3];
            float ax0 = dcx - 0.5f * dw, ay0 = dcy - 0.5f * dh;
            float ax1 = dcx + 0.5f * dw, ay1 = dcy + 0.5f * dh;
            float iw = fmaxf(fminf(ax1, bx1) - fmaxf(ax0, bx0), 0.0f);
            float ih = fmaxf(fminf(ay1, by1) - fmaxf(ay0, by0), 0.0f);
            float inter = iw * ih;
            float iou = inter / (dw * dh + areaB - inter);
            if (iou > best) { best = iou; bp = p; }
        }
        // key: larger IoU wins; equal IoU -> smaller prior index wins (first occurrence)
        unsigned long long key =
            ((unsigned long long)__float_as_uint(fmaxf(best, 0.0f)) << 32) |
            (unsigned long long)(0xFFFFFFFFu - (unsigned)bp);
        atomicMax(&sBestP[g], key);
    }
    __syncthreads();

    // ---- Forced matches: sequential over g so later g wins on duplicates ----
    if (tid == 0) {
        for (int g = 0; g < NUM_GT; ++g) {
            unsigned pu = 0xFFFFFFFFu - (unsigned)(sBestP[g] & 0xFFFFFFFFull);
            sIou[pu] = 2.0f;
            sIdx[pu] = (unsigned char)g;
        }
    }
    __syncthreads();

    // ---- Per-prior losses: CE (logsumexp), smooth-L1 on positives ----
    float posAcc = 0.0f;   // loc loss + positive CE
    int   posCnt = 0;
    const float* confB = conf + (size_t)b * NUM_PRIORS * NUM_CLASSES;
    const float* locB  = loc  + (size_t)b * NUM_PRIORS * 4;

    for (int p = tid; p < NUM_PRIORS; p += NTHREADS) {
        int  gi  = (int)sIdx[p];
        bool pos = sIou[p] > 0.5f;
        int  label = pos ? sGl[gi] : 0;
        const float* cr = confB + (size_t)p * NUM_CLASSES;
        __builtin_prefetch(cr + (size_t)NTHREADS * NUM_CLASSES, 0, 1);

        float cls[NUM_CLASSES];
        #pragma unroll
        for (int j = 0; j < NUM_CLASSES; ++j) cls[j] = cr[j];

        float m = cls[0];
        #pragma unroll
        for (int j = 1; j < NUM_CLASSES; ++j) m = fmaxf(m, cls[j]);
        float s = 0.0f;
        #pragma unroll
        for (int j = 0; j < NUM_CLASSES; ++j) s += __expf(cls[j] - m);
        float ce = m + __logf(s) - cls[label];

        float lossc;
        if (pos) {
            posCnt++;
            float bx0 = sGt[gi * 4 + 0], by0 = sGt[gi * 4 + 1];
            float bx1 = sGt[gi * 4 + 2], by1 = sGt[gi * 4 + 3];
            float gcx = 0.5f * (bx0 + bx1), gcy = 0.5f * (by0 + by1);
            float gw = bx1 - bx0, gh = by1 - by0;
            float dcx = defbox[p * 4 + 0], dcy = defbox[p * 4 + 1];
            float dw  = defbox[p * 4 + 2], dh  = defbox[p * 4 + 3];
            float e0 = (gcx - dcx) / (0.1f * dw);
            float e1 = (gcy - dcy) / (0.1f * dh);
            float e2 = __logf(gw / dw) / 0.2f;
            float e3 = __logf(gh / dh) / 0.2f;
            const float* lr = locB + (size_t)p * 4;
            posAcc += sl1(lr[0] - e0) + sl1(lr[1] - e1) +
                      sl1(lr[2] - e2) + sl1(lr[3] - e3) + ce;
            lossc = 0.0f;
        } else {
            lossc = ce;
        }
        sNeg[p] = lossc;
    }

    // ---- Fixed-order tree reduction (deterministic) ----
    sRedF[tid] = posAcc;
    sRedI[tid] = posCnt;
    __syncthreads();
    for (int s = NTHREADS / 2; s > 0; s >>= 1) {
        if (tid < s) { sRedF[tid] += sRedF[tid + s]; sRedI[tid] += sRedI[tid + s]; }
        __syncthreads();
    }
    float posLoss = sRedF[0];
    int   numPos  = sRedI[0];
    int   negCap  = NUM_PRIORS - numPos;
    int   numNeg  = 3 * numPos; if (numNeg > negCap) numNeg = negCap;
    __syncthreads();

    // ---- Exact top-k sum via 4-level radix select on float bit patterns ----
    // All values >= 0 so uint compare == float compare. Sum-of-top-k is
    // tie-invariant, matching the reference's argsort-rank mining exactly.
    float negSum = 0.0f;
    if (numNeg > 0) {                      // uniform branch (barriers OK)
        if (tid == 0) { sPrefix = 0u; sRemain = numNeg; }
        __syncthreads();
        #pragma unroll
        for (int L = 0; L < 4; ++L) {
            const int shift = 24 - 8 * L;
            for (int i = tid; i < 256; i += NTHREADS) sHist[i] = 0;
            __syncthreads();
            unsigned pref = sPrefix;
            for (int p = tid; p < NUM_PRIORS; p += NTHREADS) {
                unsigned key = __float_as_uint(sNeg[p]);
                // match on the already-fixed high bytes (u64 shift: safe at L=0)
                if (((unsigned long long)key  >> (shift + 8)) ==
                    ((unsigned long long)pref >> (shift + 8))) {
                    atomicAdd(&sHist[(key >> shift) & 0xFF], 1);
                }
            }
            __syncthreads();
            if (tid == 0) {
                int rem = sRemain, cum = 0, bsel = 0;
                for (int bin = 255; bin >= 0; --bin) {
                    int c = sHist[bin];
                    if (cum + c >= rem) { bsel = bin; break; }
                    cum += c;
                }
                sPrefix = pref | ((unsigned)bsel << shift);
                sRemain = rem - cum;   // k - count(keys with top bytes > prefix)
            }
            __syncthreads();
        }
        unsigned T  = sPrefix;          // exact k-th largest key
        int     rem = sRemain;          // = numNeg - count(key > T), >= 1
        float   tv  = __uint_as_float(T);

        float selAcc = 0.0f;
        for (int p = tid; p < NUM_PRIORS; p += NTHREADS) {
            float v = sNeg[p];
            if (__float_as_uint(v) > T) selAcc += v;
        }
        sRedF[tid] = selAcc;
        __syncthreads();
        for (int s = NTHREADS / 2; s > 0; s >>= 1) {
            if (tid < s) sRedF[tid] += sRedF[tid + s];
            __syncthreads();
        }
        negSum = sRedF[0] + (float)rem * tv;
    }

    if (tid == 0) {
        wsLoss[b] = posLoss + negSum;
        wsNpos[b] = numPos;
    }
}

// Final 128-way batch reduction using V_WMMA_F32_16X16X4_F32 with a ones B-matrix:
// D = A x Ones  => sum over all 64 A elements appears in column 0 (lanes 0 & 16).
__global__ void ssd_finalize_kernel(const float* __restrict__ wsLoss,
                                    const int*   __restrict__ wsNpos,
                                    float* __restrict__ out)
{
    int lane = threadIdx.x;  // 32 threads, one wave, EXEC all ones (WMMA requirement)
    v2f a0, a1, bones;
    a0.x = wsLoss[lane * 2 + 0];
    a0.y = wsLoss[lane * 2 + 1];
    a1.x = wsLoss[64 + lane * 2 + 0];
    a1.y = wsLoss[64 + lane * 2 + 1];
    bones.x = 1.0f; bones.y = 1.0f;
    v8f c = {};
    c = __builtin_amdgcn_wmma_f32_16x16x4_f32(false, a0, false, bones,
                                              (short)0, c, false, false);
    c = __builtin_amdgcn_wmma_f32_16x16x4_f32(false, a1, false, bones,
                                              (short)0, c, false, false);
    float s = c[0] + c[1] + c[2] + c[3] + c[4] + c[5] + c[6] + c[7];
    float total = __shfl(s, 0, 32) + __shfl(s, 16, 32);  // column N=0: rows 0..7 + 8..15
    if (lane == 0) {
        long long np = 0;
        for (int i = 0; i < BATCH; ++i) np += (long long)wsNpos[i];
        float denom = (float)(np > 1 ? np : 1);
        out[0] = total / denom;
    }
}

extern "C" void kernel_launch(void* const* d_in, const int* in_sizes, int n_in,
                              void* d_out, int out_size, void* d_ws, size_t ws_size,
                              hipStream_t stream) {
    const float* loc       = (const float*)d_in[0];
    const float* conf      = (const float*)d_in[1];
    const float* defbox    = (const float*)d_in[2];
    const float* gt_boxes  = (const float*)d_in[3];
    const int*   gt_labels = (const int*)d_in[4];

    float* wsLoss = (float*)d_ws;
    int*   wsNpos = (int*)((char*)d_ws + BATCH * sizeof(float));

    ssd_match_kernel<<<BATCH, NTHREADS, 0, stream>>>(
        loc, conf, defbox, gt_boxes, gt_labels, wsLoss, wsNpos);
    ssd_finalize_kernel<<<1, 32, 0, stream>>>(wsLoss, wsNpos, (float*)d_out);
}